// Model1_54631984005477
// MI455X (gfx1250) — compile-verified
//
#include <hip/hip_runtime.h>
#include <hip/hip_bf16.h>
#include <math.h>

// ---------------------------------------------------------------------------
// GCN link predictor for MI455X (gfx1250, wave32).
//   3x GCNConv (fp32 WMMA 16x16x4 GEMM + edge-parallel L2 atomic aggregation)
//   + MLP head (LDS-staged weights) + pred-edge gather.
// ---------------------------------------------------------------------------

typedef __attribute__((ext_vector_type(2))) float v2f;
typedef __attribute__((ext_vector_type(8))) float v8f;

static __device__ __forceinline__ int gtid() {
  return blockIdx.x * blockDim.x + threadIdx.x;
}

// ---------------- degree / normalization ----------------------------------

__global__ void k_deg_init(float* deg, int n) {
  int t = gtid();
  if (t < n) deg[t] = 1.0f;           // +1 self loop
}

__global__ void k_deg_count(const int* __restrict__ dst, float* deg, int e) {
  int t = gtid();
  if (t < e) atomicAdd(&deg[dst[t]], 1.0f);
}

__global__ void k_deg_rsqrt(float* deg, int n) {
  int t = gtid();
  if (t < n) deg[t] = rsqrtf(deg[t]);
}

// ---------------- fp32 WMMA GEMM:  H = X(nrows x DIN) @ W(DIN x DOUT) ------
// One wave computes one 16x16 output tile with V_WMMA_F32_16X16X4_F32,
// looping K in steps of 4.  Fragment layouts per CDNA5 ISA 7.12.2:
//   A 16x4:  VGPR0/1, lanes 0-15 -> K = k,k+1 ; lanes 16-31 -> K = k+2,k+3
//   B 4x16:  mirrored (rows striped across lanes)
//   C/D:     VGPR v, lane L -> row v + 8*(L>=16), col L%16
template <int DIN, int DOUT>
__global__ void k_gemm_wmma(const float* __restrict__ X,
                            const float* __restrict__ W,
                            float* __restrict__ H, int nrows) {
  const int wave = (blockIdx.x * blockDim.x + threadIdx.x) >> 5;
  const int lane = threadIdx.x & 31;
  const int ntn  = DOUT / 16;
  const int mtiles = nrows / 16;      // nrows divisible by 16 (50000 = 16*3125)
  const int mt = wave / ntn;
  const int nt = wave % ntn;
  if (mt >= mtiles) return;           // whole-wave uniform: EXEC stays all-1s

  const int m0   = mt * 16;
  const int n0   = nt * 16;
  const int half = lane >> 4;         // 0: K pair (k,k+1)  1: K pair (k+2,k+3)
  const int l16  = lane & 15;

  const float* __restrict__ xrow = X + (size_t)(m0 + l16) * DIN;

  v8f c = {};
#pragma unroll 4
  for (int k = 0; k < DIN; k += 4) {
    v2f a, b;
    a.x = xrow[k + 2 * half];
    a.y = xrow[k + 2 * half + 1];
    b.x = W[(size_t)(k + 2 * half) * DOUT + n0 + l16];
    b.y = W[(size_t)(k + 2 * half + 1) * DOUT + n0 + l16];
    c = __builtin_amdgcn_wmma_f32_16x16x4_f32(false, a, false, b,
                                              (short)0, c, false, false);
  }

#pragma unroll
  for (int v = 0; v < 8; ++v) {
    const int row = m0 + v + 8 * half;
    H[(size_t)row * DOUT + n0 + l16] = c[v];
  }
}

// ---------------- aggregation --------------------------------------------

// OUT[i,:] = H[i,:] * dis[i]^2   (self-loop term; also zero-initializes OUT)
template <int DOUT>
__global__ void k_self_init(const float* __restrict__ H,
                            const float* __restrict__ dis,
                            float* __restrict__ OUT, int n) {
  constexpr int CH = DOUT / 4;
  int t = gtid();
  int node = t / CH, c = t % CH;
  if (node >= n) return;
  float d = dis[node];
  float s = d * d;
  float4 v = ((const float4*)(H + (size_t)node * DOUT))[c];
  v.x *= s; v.y *= s; v.z *= s; v.w *= s;
  ((float4*)(OUT + (size_t)node * DOUT))[c] = v;
}

// OUT[dst,:] += H[src,:] * dis[src]*dis[dst]   (edge-parallel, float4 chunks)
template <int DOUT>
__global__ void k_edge_agg(const float* __restrict__ H,
                           const float* __restrict__ dis,
                           const int* __restrict__ src,
                           const int* __restrict__ dst,
                           float* __restrict__ OUT, int e) {
  constexpr int CH = DOUT / 4;
  int t = gtid();
  int ei = t / CH, c = t % CH;
  if (ei >= e) return;
  int s = src[ei], d = dst[ei];
  float w = dis[s] * dis[d];
  float4 v = ((const float4*)(H + (size_t)s * DOUT))[c];
  float* o = OUT + (size_t)d * DOUT + c * 4;
  atomicAdd(o + 0, v.x * w);
  atomicAdd(o + 1, v.y * w);
  atomicAdd(o + 2, v.z * w);
  atomicAdd(o + 3, v.w * w);
}

// OUT = relu(OUT + bias)
template <int DOUT>
__global__ void k_bias_relu(float* OUT, const float* __restrict__ b, int n) {
  int t = gtid();
  if (t >= n * DOUT) return;
  OUT[t] = fmaxf(OUT[t] + b[t % DOUT], 0.0f);
}

// ---------------- MLP head: 32 -> 16 -> 8 -> 1 (sigmoid) ------------------

__global__ void k_mlp_head(const float* __restrict__ X,   // n x 32
                           const float* __restrict__ lw0, const float* __restrict__ lb0,
                           const float* __restrict__ lw1, const float* __restrict__ lb1,
                           const float* __restrict__ lw2, const float* __restrict__ lb2,
                           float* __restrict__ S, int n) {
  __shared__ float w0[32 * 16], b0[16], w1[16 * 8], b1[8], w2[8], b2s;
  int t = threadIdx.x;
  for (int i = t; i < 32 * 16; i += blockDim.x) w0[i] = lw0[i];
  for (int i = t; i < 16;      i += blockDim.x) b0[i] = lb0[i];
  for (int i = t; i < 16 * 8;  i += blockDim.x) w1[i] = lw1[i];
  for (int i = t; i < 8;       i += blockDim.x) b1[i] = lb1[i];
  for (int i = t; i < 8;       i += blockDim.x) w2[i] = lw2[i];
  if (t == 0) b2s = lb2[0];
  __syncthreads();

  int node = blockIdx.x * blockDim.x + t;
  if (node >= n) return;

  float xin[32];
  const float4* xr = (const float4*)(X + (size_t)node * 32);
#pragma unroll
  for (int c = 0; c < 8; ++c) {
    float4 v = xr[c];
    xin[c * 4 + 0] = v.x; xin[c * 4 + 1] = v.y;
    xin[c * 4 + 2] = v.z; xin[c * 4 + 3] = v.w;
  }

  float h0[16];
#pragma unroll
  for (int j = 0; j < 16; ++j) {
    float acc = b0[j];
    for (int i = 0; i < 32; ++i) acc = fmaf(xin[i], w0[i * 16 + j], acc);
    h0[j] = fmaxf(acc, 0.0f);
  }
  float h1[8];
#pragma unroll
  for (int j = 0; j < 8; ++j) {
    float acc = b1[j];
    for (int i = 0; i < 16; ++i) acc = fmaf(h0[i], w1[i * 8 + j], acc);
    h1[j] = fmaxf(acc, 0.0f);
  }
  float z = b2s;
#pragma unroll
  for (int i = 0; i < 8; ++i) z = fmaf(h1[i], w2[i], z);
  S[node] = 1.0f / (1.0f + expf(-z));
}

// ---------------- pred-edge product ---------------------------------------

__global__ void k_pred(const float* __restrict__ S,
                       const int* __restrict__ pe,  // (EP,2) row-major
                       float* __restrict__ out, int ep) {
  int t = gtid();
  if (t < ep) out[t] = S[pe[2 * t]] * S[pe[2 * t + 1]];
}

// ---------------------------------------------------------------------------

static inline int cdiv(long long a, int b) { return (int)((a + b - 1) / b); }

extern "C" void kernel_launch(void* const* d_in, const int* in_sizes, int n_in,
                              void* d_out, int out_size, void* d_ws, size_t ws_size,
                              hipStream_t stream) {
  const float* x    = (const float*)d_in[0];
  const int*   ei   = (const int*)d_in[1];      // (2,E): [0..E) = src, [E..2E) = dst
  const int*   pe   = (const int*)d_in[2];      // (EP,2)
  const float* cw0  = (const float*)d_in[3];
  const float* cb0  = (const float*)d_in[4];
  const float* cw1  = (const float*)d_in[5];
  const float* cb1  = (const float*)d_in[6];
  const float* cw2  = (const float*)d_in[7];
  const float* cb2  = (const float*)d_in[8];
  const float* lw0  = (const float*)d_in[9];
  const float* lb0  = (const float*)d_in[10];
  const float* lw1  = (const float*)d_in[11];
  const float* lb1  = (const float*)d_in[12];
  const float* lw2  = (const float*)d_in[13];
  const float* lb2  = (const float*)d_in[14];

  const int N  = in_sizes[0] / 128;   // 50000
  const int E  = in_sizes[1] / 2;     // 800000
  const int EP = in_sizes[2] / 2;     // 200000
  const int* src = ei;
  const int* dst = ei + E;

  // workspace carve (256B aligned)
  char* ws = (char*)d_ws;
  size_t off = 0;
  auto carve = [&](size_t bytes) {
    void* p = ws + off;
    off += (bytes + 255) & ~(size_t)255;
    return p;
  };
  float* bufA = (float*)carve((size_t)N * 128 * sizeof(float)); // h
  float* bufB = (float*)carve((size_t)N * 128 * sizeof(float)); // layer out
  float* dis  = (float*)carve((size_t)N * sizeof(float));
  float* svec = (float*)carve((size_t)N * sizeof(float));
  (void)ws_size;

  const int BLK = 256;

  // --- normalization vector: dis = rsqrt(deg_dst + 1) ---
  k_deg_init<<<cdiv(N, BLK), BLK, 0, stream>>>(dis, N);
  k_deg_count<<<cdiv(E, BLK), BLK, 0, stream>>>(dst, dis, E);
  k_deg_rsqrt<<<cdiv(N, BLK), BLK, 0, stream>>>(dis, N);

  const int mtiles = N / 16;

  // --- layer 1: 128 -> 128 ---
  {
    long long waves = (long long)mtiles * (128 / 16);
    k_gemm_wmma<128, 128><<<cdiv(waves * 32, BLK), BLK, 0, stream>>>(x, cw0, bufA, N);
    k_self_init<128><<<cdiv((long long)N * 32, BLK), BLK, 0, stream>>>(bufA, dis, bufB, N);
    k_edge_agg<128><<<cdiv((long long)E * 32, BLK), BLK, 0, stream>>>(bufA, dis, src, dst, bufB, E);
    k_bias_relu<128><<<cdiv((long long)N * 128, BLK), BLK, 0, stream>>>(bufB, cb0, N);
  }
  // --- layer 2: 128 -> 64 ---
  {
    long long waves = (long long)mtiles * (64 / 16);
    k_gemm_wmma<128, 64><<<cdiv(waves * 32, BLK), BLK, 0, stream>>>(bufB, cw1, bufA, N);
    k_self_init<64><<<cdiv((long long)N * 16, BLK), BLK, 0, stream>>>(bufA, dis, bufB, N);
    k_edge_agg<64><<<cdiv((long long)E * 16, BLK), BLK, 0, stream>>>(bufA, dis, src, dst, bufB, E);
    k_bias_relu<64><<<cdiv((long long)N * 64, BLK), BLK, 0, stream>>>(bufB, cb1, N);
  }
  // --- layer 3: 64 -> 32 ---
  {
    long long waves = (long long)mtiles * (32 / 16);
    k_gemm_wmma<64, 32><<<cdiv(waves * 32, BLK), BLK, 0, stream>>>(bufB, cw2, bufA, N);
    k_self_init<32><<<cdiv((long long)N * 8, BLK), BLK, 0, stream>>>(bufA, dis, bufB, N);
    k_edge_agg<32><<<cdiv((long long)E * 8, BLK), BLK, 0, stream>>>(bufA, dis, src, dst, bufB, E);
    k_bias_relu<32><<<cdiv((long long)N * 32, BLK), BLK, 0, stream>>>(bufB, cb2, N);
  }

  // --- MLP head + sigmoid ---
  k_mlp_head<<<cdiv(N, BLK), BLK, 0, stream>>>(bufB, lw0, lb0, lw1, lb1, lw2, lb2, svec, N);

  // --- prediction edge scores ---
  k_pred<<<cdiv(EP, BLK), BLK, 0, stream>>>(svec, pe, (float*)d_out, EP);
}